// MultiFrameBoxLoss_32633161515881
// MI455X (gfx1250) — compile-verified
//
#include <hip/hip_runtime.h>
#include <cstdint>

// Problem constants from the reference setup: B=8, F=8 -> 64 frames,
// A=32768 anchors, O=50 truths/frame, 2 classes.
#define BF_FRAMES 64
#define A_CNT     32768
#define O_CNT     50
#define TILE      1024
#define NTHREADS  1024

#ifndef __has_builtin
#define __has_builtin(x) 0
#endif

#if __has_builtin(__builtin_amdgcn_wmma_f32_16x16x4_f32)
#define HAVE_WMMA_F32 1
#else
#define HAVE_WMMA_F32 0
#endif

#if __has_builtin(__builtin_amdgcn_global_load_async_to_lds_b128)
#define HAVE_ASYNC_LDS 1
#else
#define HAVE_ASYNC_LDS 0
#endif

#if __has_builtin(__builtin_amdgcn_s_wait_asynccnt)
#define HAVE_WAIT_ASYNC 1
#else
#define HAVE_WAIT_ASYNC 0
#endif

// Device-pass feature confirmation (shows up in compiler diagnostics).
#if defined(__gfx1250__)
#if HAVE_ASYNC_LDS
#pragma message("gfx1250: global_load_async_to_lds_b128 ACTIVE")
#else
#pragma message("gfx1250: async-to-LDS builtin NOT available (fallback copies)")
#endif
#if HAVE_WMMA_F32
#pragma message("gfx1250: wmma_f32_16x16x4_f32 ACTIVE")
#endif
#endif

typedef float v2f __attribute__((ext_vector_type(2)));
typedef float v8f __attribute__((ext_vector_type(8)));
typedef int   v4i __attribute__((ext_vector_type(4)));

// gfx1250 is wave32-only; all lane math below assumes 32 lanes.
__device__ __forceinline__ unsigned long long shflxor_u64(unsigned long long v, int m) {
  unsigned lo = (unsigned)v;
  unsigned hi = (unsigned)(v >> 32);
  lo = __shfl_xor(lo, m, 32);
  hi = __shfl_xor(hi, m, 32);
  return (((unsigned long long)hi) << 32) | lo;
}

// Stage 16 bytes/thread from global into LDS via the CDNA5 async
// global->LDS path (ASYNCcnt-tracked); fallback is a plain b128 copy.
// Builtin signature: (int4 AS1*, int4 AS3*, imm offset, imm cpol).
__device__ __forceinline__ void stage16(const float* gsrc, float* ldst) {
#if HAVE_ASYNC_LDS
  __builtin_amdgcn_global_load_async_to_lds_b128(
      (v4i __attribute__((address_space(1)))*)gsrc,
      (v4i __attribute__((address_space(3)))*)ldst, 0, 0);
#if HAVE_WAIT_ASYNC
  __builtin_amdgcn_s_wait_asynccnt(0);
#else
  asm volatile("s_wait_asynccnt 0" ::: "memory");
#endif
#else
  *(float4*)ldst = *(const float4*)gsrc;
#endif
}

// Block-wide f32 sum for 1024 threads (32 waves). Wave-level shuffle fold,
// then the 32 wave partials are combined on the matrix pipe:
// v_wmma_f32_16x16x4_f32 with an all-ones A matrix computes
// colsum[n] = p[n] + p[n+16] (K-dim + lane-half fold), freeing VALU.
// Result valid on tid==0 only.
__device__ __forceinline__ float block_reduce_sum(float v, float* s_part,
                                                  float* s_row, int tid) {
  __syncthreads();
  for (int m = 16; m; m >>= 1) v += __shfl_xor(v, m, 32);
  const int lane = tid & 31, wv = tid >> 5;
  if (lane == 0) s_part[wv] = v;
  __syncthreads();
  if (wv == 0) {
#if HAVE_WMMA_F32
    v2f a; a[0] = 1.0f; a[1] = 1.0f;          // ones 16x4 A matrix
    v2f b; b[0] = s_part[lane]; b[1] = 0.0f;  // partials as 4x16 B matrix
    v8f c = {};
    v8f d = __builtin_amdgcn_wmma_f32_16x16x4_f32(
        /*neg_a=*/false, a, /*neg_b=*/false, b,
        /*c_mod=*/(short)0, c, /*reuse_a=*/false, /*reuse_b=*/false);
    if (lane < 16) s_row[lane] = d[0];        // D row 0 = column sums
#else
    if (lane < 16) s_row[lane] = s_part[lane] + s_part[lane + 16];
#endif
  }
  __syncthreads();
  float tot = 0.0f;
  if (tid == 0)
    for (int i = 0; i < 16; ++i) tot += s_row[i];
  return tot;
}

__device__ __forceinline__ int block_reduce_sum_i(int v, int* s_parti, int tid) {
  __syncthreads();
  for (int m = 16; m; m >>= 1) v += __shfl_xor(v, m, 32);
  if ((tid & 31) == 0) s_parti[tid >> 5] = v;
  __syncthreads();
  int tot = 0;
  if (tid == 0)
    for (int i = 0; i < 32; ++i) tot += s_parti[i];
  return tot;
}

// IoU of a point-form anchor vs a point-form truth box.
__device__ __forceinline__ float iou_pf(float4 tb, float tarea,
                                        float x1, float y1, float x2, float y2,
                                        float aarea) {
  float ix = fminf(tb.z, x2) - fmaxf(tb.x, x1);
  float iy = fminf(tb.w, y2) - fmaxf(tb.y, y1);
  ix = fmaxf(ix, 0.0f);
  iy = fmaxf(iy, 0.0f);
  const float inter = ix * iy;
  return inter / (tarea + aarea - inter);
}

// ---------------------------------------------------------------------------
// Kernel 1: one block per frame. Match anchors<->truths, force-match, compute
// smooth-L1 loc loss (pos only), per-anchor CE, positive-CE sum, num_pos, and
// the hard-negative-mining array hnm = pos ? 0 : ce.
//
// Per-truth argmax over anchors is register-local: wave w exclusively owns
// truths w and w+32, keeps a packed (iou_bits<<32)|~a running max across all
// tiles, and does ONE 5-step wave fold per truth at the end (no LDS atomics,
// no per-iteration ds_bpermute storm). ~a => ties resolve to smallest anchor
// index, matching jnp.argmax first-occurrence semantics.
// ---------------------------------------------------------------------------
__global__ __launch_bounds__(NTHREADS)
void mfbl_match_kernel(const float* __restrict__ loc,
                       const float* __restrict__ conf,
                       const float* __restrict__ anchors,
                       const float* __restrict__ targets,
                       float* __restrict__ hnm,
                       float* __restrict__ frameL,
                       float* __restrict__ posce,
                       int* __restrict__ npos) {
  const int bf   = blockIdx.x;
  const int tid  = threadIdx.x;
  const int lane = tid & 31;
  const int wv   = tid >> 5;

  __shared__ __align__(16) float4 s_tbox[O_CNT];   // truth xyxy
  __shared__ float s_tarea[O_CNT], s_tlab[O_CNT];
  __shared__ unsigned long long s_best[O_CNT];     // packed (iou_bits<<32)|~a
  __shared__ unsigned char s_abest[A_CNT];         // bits[5:0]=truth idx, bit6=pos
  __shared__ __align__(16) float s_tile[TILE * 4]; // anchor tile (cx,cy,w,h)
  __shared__ float s_part[32], s_row[16];
  __shared__ int s_parti[32];

  // Load truths for this frame; point form + area.
  if (tid < O_CNT) {
    const float* t = targets + ((size_t)bf * O_CNT + tid) * 5;
    float4 b;
    b.x = t[0]; b.y = t[1]; b.z = t[2]; b.w = t[3];
    s_tbox[tid]  = b;
    s_tarea[tid] = (b.z - b.x) * (b.w - b.y);
    s_tlab[tid]  = t[4];
  }
  __syncthreads();

  // This wave's owned truths, hoisted into registers.
  const int tA = wv;                 // 0..31, always < O_CNT
  const int tB = wv + 32;            // valid while < O_CNT
  const bool hasB = (tB < O_CNT);
  const float4 boxA = s_tbox[tA];
  const float  areaA = s_tarea[tA];
  float4 boxB = boxA;
  float  areaB = areaA;
  if (hasB) { boxB = s_tbox[tB]; areaB = s_tarea[tB]; }
  unsigned long long keyA = 0ull, keyB = 0ull;

  // ---- Pass 1: IoU; per-anchor argmax over truths; per-truth running max.
  for (int tile = 0; tile < A_CNT / TILE; ++tile) {
    __syncthreads();
    stage16(anchors + (size_t)(tile * TILE + tid) * 4, &s_tile[tid * 4]);
    __syncthreads();

    // (a) per-anchor best truth (first-max, like argmax axis=0)
    const int a = tile * TILE + tid;
    const float4 av = *(const float4*)&s_tile[tid * 4];
    const float ax1 = av.x - 0.5f * av.z, ay1 = av.y - 0.5f * av.w;
    const float ax2 = av.x + 0.5f * av.z, ay2 = av.y + 0.5f * av.w;
    const float aarea = av.z * av.w;
    float bestI = -1.0f;
    int   bestO = 0;
    for (int o = 0; o < O_CNT; ++o) {
      const float iou = iou_pf(s_tbox[o], s_tarea[o], ax1, ay1, ax2, ay2, aarea);
      if (iou > bestI) { bestI = iou; bestO = o; }
    }
    s_abest[a] = (unsigned char)(bestO | (bestI >= 0.5f ? 0x40 : 0x00));

    // (b) per-truth best anchor over this tile (register-local running max)
    for (int j = lane; j < TILE; j += 32) {
      const float4 cv = *(const float4*)&s_tile[j * 4];
      const float cx1 = cv.x - 0.5f * cv.z, cy1 = cv.y - 0.5f * cv.w;
      const float cx2 = cv.x + 0.5f * cv.z, cy2 = cv.y + 0.5f * cv.w;
      const float carea = cv.z * cv.w;
      const unsigned aj = (unsigned)(tile * TILE + j);
      {
        const float iou = iou_pf(boxA, areaA, cx1, cy1, cx2, cy2, carea);
        const unsigned long long key =
            (((unsigned long long)__float_as_uint(iou)) << 32) | (~aj);
        if (key > keyA) keyA = key;
      }
      if (hasB) {
        const float iou = iou_pf(boxB, areaB, cx1, cy1, cx2, cy2, carea);
        const unsigned long long key =
            (((unsigned long long)__float_as_uint(iou)) << 32) | (~aj);
        if (key > keyB) keyB = key;
      }
    }
  }

  // One fold per owned truth; exclusive owner writes, no atomics needed.
  for (int m = 16; m; m >>= 1) {
    unsigned long long o2 = shflxor_u64(keyA, m);
    if (o2 > keyA) keyA = o2;
    o2 = shflxor_u64(keyB, m);
    if (o2 > keyB) keyB = o2;
  }
  if (lane == 0) {
    s_best[tA] = keyA;
    if (hasB) s_best[tB] = keyB;
  }
  __syncthreads();

  // Force-match: each truth claims its best anchor; serial => last-wins for
  // duplicate best anchors (scatter semantics).
  if (tid == 0) {
    for (int t = 0; t < O_CNT; ++t) {
      const unsigned aa = ~(unsigned)s_best[t];  // recover anchor index
      s_abest[aa] = (unsigned char)(t | 0x40);
    }
  }
  __syncthreads();

  // ---- Pass 2: encode + smooth-L1 (pos only), CE, hnm array.
  float locsum = 0.0f, pcesum = 0.0f;
  int npcnt = 0;
  const size_t fbase = (size_t)bf * A_CNT;
  for (int tile = 0; tile < A_CNT / TILE; ++tile) {
    __syncthreads();
    stage16(anchors + (size_t)(tile * TILE + tid) * 4, &s_tile[tid * 4]);
    __syncthreads();

    const int a = tile * TILE + tid;
    if (tile + 1 < A_CNT / TILE) {  // prefetch next tile's loc stream
      __builtin_prefetch(loc + (fbase + (size_t)a + TILE) * 4, 0, 1);
    }
    const float4 av = *(const float4*)&s_tile[tid * 4];

    const unsigned e = s_abest[a];
    const int  ti  = e & 63;
    const bool pos = (e & 0x40) != 0;
    const float lab = s_tlab[ti];
    const bool p = pos && (lab > 0.0f);

    if (p) {
      const float4 tb = s_tbox[ti];
      const float mcx = 0.5f * (tb.x + tb.z), mcy = 0.5f * (tb.y + tb.w);
      const float mw = tb.z - tb.x, mh = tb.w - tb.y;
      const float g0 = (mcx - av.x) / (0.1f * av.z);
      const float g1 = (mcy - av.y) / (0.1f * av.w);
      const float g2 = logf(mw / av.z) * 5.0f;   // /0.2
      const float g3 = logf(mh / av.w) * 5.0f;
      const float4 lp = *(const float4*)(loc + (fbase + (size_t)a) * 4);
      const float d0 = fabsf(lp.x - g0), d1 = fabsf(lp.y - g1);
      const float d2 = fabsf(lp.z - g2), d3 = fabsf(lp.w - g3);
      locsum += (d0 < 1.0f ? 0.5f * d0 * d0 : d0 - 0.5f)
              + (d1 < 1.0f ? 0.5f * d1 * d1 : d1 - 0.5f)
              + (d2 < 1.0f ? 0.5f * d2 * d2 : d2 - 0.5f)
              + (d3 < 1.0f ? 0.5f * d3 * d3 : d3 - 0.5f);
      npcnt += 1;
    }

    const float2 cp = *(const float2*)(conf + (fbase + (size_t)a) * 2);
    const float mx = fmaxf(cp.x, cp.y);
    const float lse = mx + logf(expf(cp.x - mx) + expf(cp.y - mx));
    const int tgt = p ? (int)lab : 0;          // labels are class 1 here
    const float ce = lse - (tgt == 1 ? cp.y : cp.x);
    hnm[fbase + (size_t)a] = p ? 0.0f : ce;
    if (p) pcesum += ce;
  }

  const float Lt = block_reduce_sum(locsum, s_part, s_row, tid);
  const float Pt = block_reduce_sum(pcesum, s_part, s_row, tid);
  const int   Nt = block_reduce_sum_i(npcnt, s_parti, tid);
  if (tid == 0) {
    frameL[bf] = Lt;
    posce[bf]  = Pt;
    npos[bf]   = Nt;
  }
}

// ---------------------------------------------------------------------------
// Kernel 2: one block per frame. Exact top-k sum of hnm via 4-pass radix
// select on non-negative f32 bit patterns (order-preserving as u32).
// loss_c[frame] = posce + sum(top-k negatives), k = min(3*num_pos, A-1).
// Ties at the threshold contribute r * threshold (sum-exact vs stable sort).
// ---------------------------------------------------------------------------
__global__ __launch_bounds__(NTHREADS)
void mfbl_mine_kernel(const float* __restrict__ hnm,
                      const float* __restrict__ posce,
                      const int* __restrict__ npos,
                      float* __restrict__ frameC) {
  const int bf = blockIdx.x;
  const int tid = threadIdx.x;
  __shared__ unsigned s_hist[256];
  __shared__ unsigned s_prefix;
  __shared__ int s_remaining;
  __shared__ float s_part[32], s_row[16];

  const size_t fbase = (size_t)bf * A_CNT;
  int k = npos[bf] * 3;
  if (k > A_CNT - 1) k = A_CNT - 1;

  float extra = 0.0f;
  if (k > 0) {
    if (tid == 0) { s_prefix = 0u; s_remaining = k; }
    for (int pass = 3; pass >= 0; --pass) {
      __syncthreads();
      if (tid < 256) s_hist[tid] = 0u;
      __syncthreads();
      const unsigned pref = s_prefix;
      for (int i = tid; i < A_CNT; i += NTHREADS) {
        const unsigned bits = __float_as_uint(hnm[fbase + i]);
        const bool match = (pass == 3) || ((bits >> ((pass + 1) * 8)) == pref);
        if (match) atomicAdd(&s_hist[(bits >> (pass * 8)) & 255u], 1u);
      }
      __syncthreads();
      if (tid == 0) {
        const int rem = s_remaining;
        unsigned cum = 0;
        int chosen = 0;
        for (int b = 255; b >= 0; --b) {
          const unsigned c = s_hist[b];
          if (cum + c >= (unsigned)rem) { chosen = b; break; }
          cum += c;
        }
        s_remaining = rem - (int)cum;
        s_prefix = (pref << 8) | (unsigned)chosen;
      }
    }
    __syncthreads();
    const unsigned tb = s_prefix;   // bit pattern of the k-th largest value
    const int r = s_remaining;      // how many at exactly tb to take
    float sum_gt = 0.0f;
    for (int i = tid; i < A_CNT; i += NTHREADS) {
      const float v = hnm[fbase + i];
      if (__float_as_uint(v) > tb) sum_gt += v;
    }
    const float tot = block_reduce_sum(sum_gt, s_part, s_row, tid);
    if (tid == 0) extra = tot + (float)r * __uint_as_float(tb);
  }
  if (tid == 0) frameC[bf] = posce[bf] + extra;
}

// ---------------------------------------------------------------------------
// Kernel 3: deterministic fixed-order final sums -> d_out[0..1].
// ---------------------------------------------------------------------------
__global__ void mfbl_final_kernel(const float* __restrict__ frameL,
                                  const float* __restrict__ frameC,
                                  float* __restrict__ out) {
  if (threadIdx.x == 0 && blockIdx.x == 0) {
    float l = 0.0f, c = 0.0f;
    for (int i = 0; i < BF_FRAMES; ++i) { l += frameL[i]; c += frameC[i]; }
    out[0] = l;
    out[1] = c;
  }
}

extern "C" void kernel_launch(void* const* d_in, const int* in_sizes, int n_in,
                              void* d_out, int out_size, void* d_ws, size_t ws_size,
                              hipStream_t stream) {
  (void)in_sizes; (void)n_in; (void)out_size; (void)ws_size;
  const float* loc     = (const float*)d_in[0];  // (8, 8*32768, 4)
  const float* conf    = (const float*)d_in[1];  // (8, 8*32768, 2)
  const float* anchors = (const float*)d_in[2];  // (32768, 4) cxcywh
  const float* targets = (const float*)d_in[3];  // (8, 8, 50, 5)

  float* hnm    = (float*)d_ws;                       // BF*A floats (8 MiB)
  float* frameL = hnm + (size_t)BF_FRAMES * A_CNT;    // 64
  float* posce  = frameL + BF_FRAMES;                 // 64
  float* frameC = posce + BF_FRAMES;                  // 64
  int*   npos   = (int*)(frameC + BF_FRAMES);         // 64

  float* out = (float*)d_out;  // [loss_l, loss_c]

  mfbl_match_kernel<<<BF_FRAMES, NTHREADS, 0, stream>>>(
      loc, conf, anchors, targets, hnm, frameL, posce, npos);
  mfbl_mine_kernel<<<BF_FRAMES, NTHREADS, 0, stream>>>(hnm, posce, npos, frameC);
  mfbl_final_kernel<<<1, 64, 0, stream>>>(frameL, frameC, out);
}